// DGCNN_MODULAR_63256278336271
// MI455X (gfx1250) — compile-verified
//
#include <hip/hip_runtime.h>
#include <hip/hip_bf16.h>
#include <math.h>

typedef __attribute__((ext_vector_type(2))) float v2f;
typedef __attribute__((ext_vector_type(8))) float v8f;

#define BATCH 4
#define NPTS  2048
#define KNN   20
#define CATC  960
#define EPS   1e-5f

__device__ __forceinline__ float lrelu(float v) { return v > 0.f ? v : 0.2f * v; }

__device__ __forceinline__ v8f wmma4(v2f a, v2f b, v8f c) {
    // V_WMMA_F32_16X16X4_F32 : D(16x16,f32) = A(16x4,f32) * B(4x16,f32) + C
    return __builtin_amdgcn_wmma_f32_16x16x4_f32(false, a, false, b, (short)0, c, false, false);
}

// ---------------------------------------------------------------------------
// Layer 0: edge dim 6 -> 64 channels. Tiny K, plain VALU.
// block (64 co, 4 points). Writes raw k-max (b,64,n) + channel sum/sumsq.
// ---------------------------------------------------------------------------
__global__ void layer0_kernel(const float* __restrict__ x, const int* __restrict__ idx,
                              const float* __restrict__ w0, float* __restrict__ raw,
                              float* __restrict__ stats) {
    int co = threadIdx.x, pp = threadIdx.y;
    int p = blockIdx.x * 4 + pp;
    int b = p >> 11, n = p & (NPTS - 1);
    const float* xc = x + ((size_t)(b * NPTS + n)) * 3;
    float cx = xc[0], cy = xc[1], cz = xc[2];
    float w[6];
#pragma unroll
    for (int j = 0; j < 6; ++j) w[j] = w0[co * 6 + j];
    const int* ip = idx + ((size_t)(b * NPTS + n)) * KNN;
    float mx = -3.4e38f, s = 0.f, ss = 0.f;
    for (int kk = 0; kk < KNN; ++kk) {
        const float* xn = x + ((size_t)(b * NPTS + ip[kk])) * 3;
        float y = w[0] * (xn[0] - cx) + w[1] * (xn[1] - cy) + w[2] * (xn[2] - cz)
                + w[3] * cx + w[4] * cy + w[5] * cz;
        mx = fmaxf(mx, y); s += y; ss += y * y;
    }
    raw[((size_t)(b * 64 + co)) * NPTS + n] = mx;
    __shared__ float rs[4][64], rss[4][64];
    rs[pp][co] = s; rss[pp][co] = ss;
    __syncthreads();
    if (pp == 0) {
        float S = rs[0][co] + rs[1][co] + rs[2][co] + rs[3][co];
        float SS = rss[0][co] + rss[1][co] + rss[2][co] + rss[3][co];
        atomicAdd(&stats[co], S);
        atomicAdd(&stats[1024 + co], SS);
    }
}

// ---------------------------------------------------------------------------
// EdgeConv GEMM via f32 WMMA. CIN = input channels, KD = 2*CIN = COUT.
// Grid: (B*N/8, COUT/16). Block 256 = 8 waves; wave w handles point
// blockIdx.x*8+w, output tile 16co x 32 padded-k columns (20 real).
// Fuses gather, GEMM, k-max, and per-channel sum/sumsq.
// ---------------------------------------------------------------------------
template <int CIN>
__global__ __launch_bounds__(256) void edgeconv_wmma(const float* __restrict__ in,
                                                     const int* __restrict__ idx,
                                                     const float* __restrict__ W,
                                                     float* __restrict__ raw,
                                                     float* __restrict__ stats) {
    constexpr int KD = 2 * CIN;     // edge-feature / contraction dim
    constexpr int COUT = 2 * CIN;
    __shared__ float Wl[16 * KD];
    __shared__ float Yl[8][16][32];
    __shared__ int   nbs[8][32];
    __shared__ float red[2][16][8];

    int tid = threadIdx.x;
    int wave = tid >> 5, lane = tid & 31;
    int co_base = blockIdx.y * 16;

    // Stage W tile (16 rows of co) into LDS, shared by all 8 waves.
    for (int i = tid; i < 16 * KD; i += 256)
        Wl[i] = W[(size_t)(co_base + i / KD) * KD + (i % KD)];

    int p = blockIdx.x * 8 + wave;
    int b = p >> 11, n = p & (NPTS - 1);
    nbs[wave][lane] = (lane < KNN) ? idx[((size_t)(b * NPTS + n)) * KNN + lane] : n;
    __syncthreads();

    const float* inb = in + (size_t)b * CATC * NPTS;  // batch stride = 960*N
    int half = lane >> 4;            // 0: K rows {0,1}; 1: K rows {2,3}
    int ml = lane & 15;
    int nb0 = nbs[wave][ml];
    int nb1 = nbs[wave][ml + 16];
    const float* wrow = &Wl[ml * KD];

    v8f acc[2] = {v8f{}, v8f{}};
    for (int k0 = 0; k0 < KD; k0 += 4) {
        int kb = k0 + 2 * half;
        v2f a;  a.x = wrow[kb]; a.y = wrow[kb + 1];
        v2f b0v, b1v;
        if (k0 < CIN) {  // uniform branch: CIN % 4 == 0
            float c0 = inb[(size_t)kb * NPTS + n];
            float c1 = inb[(size_t)(kb + 1) * NPTS + n];
            b0v.x = inb[(size_t)kb * NPTS + nb0] - c0;
            b0v.y = inb[(size_t)(kb + 1) * NPTS + nb0] - c1;
            b1v.x = inb[(size_t)kb * NPTS + nb1] - c0;
            b1v.y = inb[(size_t)(kb + 1) * NPTS + nb1] - c1;
        } else {         // center half of edge feature (column-independent)
            float c0 = inb[(size_t)(kb - CIN) * NPTS + n];
            float c1 = inb[(size_t)(kb + 1 - CIN) * NPTS + n];
            b0v.x = c0; b0v.y = c1; b1v.x = c0; b1v.y = c1;
        }
        acc[0] = wmma4(a, b0v, acc[0]);
        acc[1] = wmma4(a, b1v, acc[1]);
    }

    // C layout: lane L, vgpr r -> co_local = r + 8*(L>>4), col = L&15 (+16 for tile 1)
#pragma unroll
    for (int t = 0; t < 2; ++t)
#pragma unroll
        for (int r = 0; r < 8; ++r)
            Yl[wave][r + 8 * half][16 * t + ml] = acc[t][r];
    __syncthreads();

    // k-max over the 20 valid neighbors + channel statistics
    if (tid < 128) {
        int co_l = tid & 15, pp = tid >> 4;
        float mx = -3.4e38f, s = 0.f, ss = 0.f;
        for (int kk = 0; kk < KNN; ++kk) {
            float y = Yl[pp][co_l][kk];
            mx = fmaxf(mx, y); s += y; ss += y * y;
        }
        int p2 = blockIdx.x * 8 + pp;
        int b2 = p2 >> 11, n2 = p2 & (NPTS - 1);
        raw[((size_t)(b2 * COUT + co_base + co_l)) * NPTS + n2] = mx;
        red[0][co_l][pp] = s; red[1][co_l][pp] = ss;
    }
    __syncthreads();
    if (tid < 16) {
        float s = 0.f, ss = 0.f;
        for (int pp = 0; pp < 8; ++pp) { s += red[0][tid][pp]; ss += red[1][tid][pp]; }
        atomicAdd(&stats[co_base + tid], s);
        atomicAdd(&stats[1024 + co_base + tid], ss);
    }
}

// ---------------------------------------------------------------------------
// Apply BN (from accumulated stats) + LeakyReLU; write into cat buffer slot.
// raw is (B,C,N) contiguous; out goes to cat with channel stride N, batch 960*N.
// ---------------------------------------------------------------------------
__global__ void bn_apply_kernel(const float* __restrict__ raw, float* __restrict__ outb,
                                const float* __restrict__ stats, const float* __restrict__ g,
                                const float* __restrict__ bb, int C, float invcnt) {
    int i = blockIdx.x * 256 + threadIdx.x;
    int n = i & (NPTS - 1);
    int t = i >> 11;
    int c = t % C, b = t / C;
    float mean = stats[c] * invcnt;
    float var = stats[1024 + c] * invcnt - mean * mean;
    float sc = g[c] * rsqrtf(var + EPS);
    float sh = bb[c] - mean * sc;
    outb[((size_t)b * CATC + c) * NPTS + n] = lrelu(sc * raw[i] + sh);
}

// ---------------------------------------------------------------------------
// feats = w4(1024x960) @ cat. WMMA, no gather. Grid (8192/256 cols, 1024/16 co).
// Wave handles 16co x 32 cols; writes raw Y to feats area + channel stats.
// ---------------------------------------------------------------------------
__global__ __launch_bounds__(256) void feats_gemm(const float* __restrict__ cat,
                                                  const float* __restrict__ w4,
                                                  float* __restrict__ feats,
                                                  float* __restrict__ stats) {
    __shared__ float Yl[8][16][32];
    __shared__ float red[2][16][16];
    int tid = threadIdx.x, wave = tid >> 5, lane = tid & 31;
    int half = lane >> 4, ml = lane & 15;
    int co_base = blockIdx.y * 16;
    int colbase = blockIdx.x * 256 + wave * 32;        // 256-aligned -> single b
    int b = colbase >> 11;
    int n0 = (colbase & (NPTS - 1)) + ml;
    const float* catb = cat + (size_t)b * CATC * NPTS;
    const float* wrow = w4 + (size_t)(co_base + ml) * CATC;

    v8f acc[2] = {v8f{}, v8f{}};
    for (int k0 = 0; k0 < CATC; k0 += 4) {
        int kb = k0 + 2 * half;
        v2f a;  a.x = wrow[kb]; a.y = wrow[kb + 1];
        v2f b0v, b1v;
        b0v.x = catb[(size_t)kb * NPTS + n0];
        b0v.y = catb[(size_t)(kb + 1) * NPTS + n0];
        b1v.x = catb[(size_t)kb * NPTS + n0 + 16];
        b1v.y = catb[(size_t)(kb + 1) * NPTS + n0 + 16];
        acc[0] = wmma4(a, b0v, acc[0]);
        acc[1] = wmma4(a, b1v, acc[1]);
    }
#pragma unroll
    for (int t = 0; t < 2; ++t)
#pragma unroll
        for (int r = 0; r < 8; ++r) {
            int co = co_base + r + 8 * half;
            int n = (colbase & (NPTS - 1)) + 16 * t + ml;
            feats[((size_t)(b * 1024 + co)) * NPTS + n] = acc[t][r];
            Yl[wave][r + 8 * half][16 * t + ml] = acc[t][r];
        }
    __syncthreads();
    {   // per-channel sum/sumsq over this WG's 256 columns
        int co_l = tid & 15, seg = tid >> 4;   // 16 segs x 16 cols
        float s = 0.f, ss = 0.f;
        for (int j = 0; j < 16; ++j) {
            int col = seg * 16 + j;
            float y = Yl[col >> 5][co_l][col & 31];
            s += y; ss += y * y;
        }
        red[0][co_l][seg] = s; red[1][co_l][seg] = ss;
    }
    __syncthreads();
    if (tid < 16) {
        float s = 0.f, ss = 0.f;
        for (int j = 0; j < 16; ++j) { s += red[0][tid][j]; ss += red[1][tid][j]; }
        atomicAdd(&stats[co_base + tid], s);
        atomicAdd(&stats[1024 + co_base + tid], ss);
    }
}

// BN+lrelu feats in place; per-(b,c) max & mean over N. One block per (b,c).
__global__ void pool_bn_kernel(float* __restrict__ feats, const float* __restrict__ stats,
                               const float* __restrict__ g, const float* __restrict__ bb,
                               float* __restrict__ fmax, float* __restrict__ fmean) {
    int blk = blockIdx.x, b = blk >> 10, c = blk & 1023, tid = threadIdx.x;
    float mean = stats[c] * (1.f / 8192.f);
    float var = stats[1024 + c] * (1.f / 8192.f) - mean * mean;
    float sc = g[c] * rsqrtf(var + EPS);
    float sh = bb[c] - mean * sc;
    float* row = feats + ((size_t)(b * 1024 + c)) * NPTS;
    float mx = -3.4e38f, s = 0.f;
    for (int n = tid; n < NPTS; n += 256) {
        float v = lrelu(sc * row[n] + sh);
        row[n] = v; mx = fmaxf(mx, v); s += v;
    }
    __shared__ float sm[256], ss[256];
    sm[tid] = mx; ss[tid] = s; __syncthreads();
    for (int o = 128; o > 0; o >>= 1) {
        if (tid < o) { sm[tid] = fmaxf(sm[tid], sm[tid + o]); ss[tid] += ss[tid + o]; }
        __syncthreads();
    }
    if (tid == 0) { fmax[b * 1024 + c] = sm[0]; fmean[b * 1024 + c] = ss[0] * (1.f / NPTS); }
}

// ------------------------- FC head (tiny) ----------------------------------
__global__ void fc1_kernel(const float* __restrict__ lw1, const float* __restrict__ fmax,
                           const float* __restrict__ fmean, float* __restrict__ out) {
    int i = blockIdx.x * 256 + threadIdx.x;
    int b = i >> 10, o = i & 1023;
    const float* wr = lw1 + (size_t)o * 2048;
    const float* xm = fmax + b * 1024;
    const float* xa = fmean + b * 1024;
    float acc = 0.f;
    for (int c = 0; c < 1024; ++c) acc += wr[c] * xm[c];
    for (int c = 0; c < 1024; ++c) acc += wr[1024 + c] * xa[c];
    out[i] = acc;
}

__global__ void bn_batch_kernel(const float* __restrict__ raw, float* __restrict__ outp,
                                const float* __restrict__ g, const float* __restrict__ bb, int C) {
    int c = blockIdx.x * 256 + threadIdx.x;
    if (c >= C) return;
    float v0 = raw[c], v1 = raw[C + c], v2 = raw[2 * C + c], v3 = raw[3 * C + c];
    float m = 0.25f * (v0 + v1 + v2 + v3);
    float var = 0.25f * (v0 * v0 + v1 * v1 + v2 * v2 + v3 * v3) - m * m;
    float sc = g[c] * rsqrtf(var + EPS);
    float sh = bb[c] - m * sc;
    outp[c]         = lrelu(sc * v0 + sh);
    outp[C + c]     = lrelu(sc * v1 + sh);
    outp[2 * C + c] = lrelu(sc * v2 + sh);
    outp[3 * C + c] = lrelu(sc * v3 + sh);
}

__global__ void fc2_kernel(const float* __restrict__ lw2, const float* __restrict__ lb2,
                           const float* __restrict__ z1, float* __restrict__ out) {
    int i = blockIdx.x * 256 + threadIdx.x;
    int b = i >> 9, o = i & 511;
    const float* wr = lw2 + (size_t)o * 1024;
    const float* xv = z1 + b * 1024;
    float acc = lb2[o];
    for (int c = 0; c < 1024; ++c) acc += wr[c] * xv[c];
    out[i] = acc;
}

__global__ void fc3_kernel(const float* __restrict__ lw3, const float* __restrict__ lb3,
                           const float* __restrict__ z2, float* __restrict__ out) {
    int i = blockIdx.x * 256 + threadIdx.x;
    int b = i >> 8, o = i & 255;
    const float* wr = lw3 + (size_t)o * 512;
    const float* xv = z2 + b * 512;
    float acc = lb3[o];
    for (int c = 0; c < 512; ++c) acc += wr[c] * xv[c];
    out[b * 256 + o] = acc;
}

// ---------------------------------------------------------------------------
extern "C" void kernel_launch(void* const* d_in, const int* in_sizes, int n_in,
                              void* d_out, int out_size, void* d_ws, size_t ws_size,
                              hipStream_t stream) {
    const float* x   = (const float*)d_in[0];
    const int*   idx = (const int*)d_in[1];
    const float *w0 = (const float*)d_in[2],  *g0 = (const float*)d_in[3],  *b0 = (const float*)d_in[4];
    const float *w1 = (const float*)d_in[5],  *g1 = (const float*)d_in[6],  *b1 = (const float*)d_in[7];
    const float *w2 = (const float*)d_in[8],  *g2 = (const float*)d_in[9],  *b2 = (const float*)d_in[10];
    const float *w3 = (const float*)d_in[11], *g3 = (const float*)d_in[12], *b3 = (const float*)d_in[13];
    const float *w4 = (const float*)d_in[14], *g4 = (const float*)d_in[15], *b4 = (const float*)d_in[16];
    const float *lw1 = (const float*)d_in[17], *g6 = (const float*)d_in[18], *b6 = (const float*)d_in[19];
    const float *lw2 = (const float*)d_in[20], *lb2 = (const float*)d_in[21];
    const float *g7 = (const float*)d_in[22], *b7 = (const float*)d_in[23];
    const float *lw3 = (const float*)d_in[24], *lb3 = (const float*)d_in[25];

    float* z     = (float*)d_out;          // (4,256)
    float* feats = z + 1024;               // (4,1024,2048)

    float* ws    = (float*)d_ws;
    float* cat   = ws;                     // (4,960,2048)
    float* raw   = cat + (size_t)BATCH * CATC * NPTS;   // (4,512,2048) max
    float* stats = raw + (size_t)BATCH * 512 * NPTS;    // 2048: sum | sumsq
    float* fmax  = stats + 2048;           // (4,1024)
    float* fmean = fmax + 4096;
    float* z1r   = fmean + 4096;
    float* z1    = z1r + 4096;
    float* z2r   = z1 + 4096;
    float* z2    = z2r + 2048;

    const float invc_edge = 1.f / (BATCH * NPTS * KNN); // 163840
    const size_t statbytes = 2048 * sizeof(float);

    // ---- layer 0 (VALU, edge dim 6 -> 64) ----
    hipMemsetAsync(stats, 0, statbytes, stream);
    layer0_kernel<<<2048, dim3(64, 4), 0, stream>>>(x, idx, w0, raw, stats);
    bn_apply_kernel<<<(BATCH * 64 * NPTS) / 256, 256, 0, stream>>>(raw, cat, stats, g0, b0, 64, invc_edge);

    // ---- layer 1: 64 -> 128 ----
    hipMemsetAsync(stats, 0, statbytes, stream);
    edgeconv_wmma<64><<<dim3(1024, 8), 256, 0, stream>>>(cat, idx, w1, raw, stats);
    bn_apply_kernel<<<(BATCH * 128 * NPTS) / 256, 256, 0, stream>>>(raw, cat + 64 * NPTS, stats, g1, b1, 128, invc_edge);

    // ---- layer 2: 128 -> 256 ----
    hipMemsetAsync(stats, 0, statbytes, stream);
    edgeconv_wmma<128><<<dim3(1024, 16), 256, 0, stream>>>(cat + 64 * NPTS, idx, w2, raw, stats);
    bn_apply_kernel<<<(BATCH * 256 * NPTS) / 256, 256, 0, stream>>>(raw, cat + 192 * NPTS, stats, g2, b2, 256, invc_edge);

    // ---- layer 3: 256 -> 512 ----
    hipMemsetAsync(stats, 0, statbytes, stream);
    edgeconv_wmma<256><<<dim3(1024, 32), 256, 0, stream>>>(cat + 192 * NPTS, idx, w3, raw, stats);
    bn_apply_kernel<<<(BATCH * 512 * NPTS) / 256, 256, 0, stream>>>(raw, cat + 448 * NPTS, stats, g3, b3, 512, invc_edge);

    // ---- feats = lrelu(bn(w4 @ cat)) + pooling ----
    hipMemsetAsync(stats, 0, statbytes, stream);
    feats_gemm<<<dim3(32, 64), 256, 0, stream>>>(cat, w4, feats, stats);
    pool_bn_kernel<<<4096, 256, 0, stream>>>(feats, stats, g4, b4, fmax, fmean);

    // ---- FC head ----
    fc1_kernel<<<16, 256, 0, stream>>>(lw1, fmax, fmean, z1r);
    bn_batch_kernel<<<4, 256, 0, stream>>>(z1r, z1, g6, b6, 1024);
    fc2_kernel<<<8, 256, 0, stream>>>(lw2, lb2, z1, z2r);
    bn_batch_kernel<<<2, 256, 0, stream>>>(z2r, z2, g7, b7, 512);
    fc3_kernel<<<4, 256, 0, stream>>>(lw3, lb3, z2, z);
}